// LQRStepClass_8400956031303
// MI455X (gfx1250) — compile-verified
//
#include <hip/hip_runtime.h>
#include <hip/hip_bf16.h>

#define T_  32
#define B_  512
#define NS_ 64
#define NC_ 32
#define SZ_ 96   // NS + NC

#define LDC 100  // LDS row stride (floats) for C/Q and F tiles: 400B, 16B-aligned
#define LDS65 65 // LDS row stride for FV / V / W / K

typedef __attribute__((ext_vector_type(2))) float v2f;
typedef __attribute__((ext_vector_type(8))) float v8f;

// ---------------------------------------------------------------------------
// f32 WMMA helpers (V_WMMA_F32_16X16X4_F32, wave32).
// A element (m,k) at A[m*a_rs + k*a_cs]; B element (k,n) at B[k*b_rs + n*b_cs].
// ---------------------------------------------------------------------------
__device__ __forceinline__ v8f zero8() {
    v8f z = {0.f, 0.f, 0.f, 0.f, 0.f, 0.f, 0.f, 0.f};
    return z;
}

__device__ __forceinline__ v8f load_tile(const float* Cm, int ld, int lane) {
    const int half = lane >> 4;
    const int l    = lane & 15;
    v8f acc;
#pragma unroll
    for (int r = 0; r < 8; ++r) acc[r] = Cm[(r + 8 * half) * ld + l];
    return acc;
}

__device__ __forceinline__ void store_tile(float* D, int ld, int lane, v8f acc) {
    const int half = lane >> 4;
    const int l    = lane & 15;
#pragma unroll
    for (int r = 0; r < 8; ++r) D[(r + 8 * half) * ld + l] = acc[r];
}

__device__ __forceinline__ v8f wmma_gemm_tile(const float* A, int a_rs, int a_cs,
                                              const float* Bm, int b_rs, int b_cs,
                                              int kdim, v8f acc, int lane) {
    const int half = lane >> 4;
    const int l    = lane & 15;
    for (int kk = 0; kk < kdim; kk += 4) {
        const int k0 = kk + 2 * half;
        v2f a, b;
        a.x = A[l * a_rs + k0 * a_cs];
        a.y = A[l * a_rs + (k0 + 1) * a_cs];
        b.x = Bm[k0 * b_rs + l * b_cs];
        b.y = Bm[(k0 + 1) * b_rs + l * b_cs];
        acc = __builtin_amdgcn_wmma_f32_16x16x4_f32(false, a, false, b,
                                                    (short)0, acc, false, false);
    }
    return acc;
}

// ---------------------------------------------------------------------------
// Async global->LDS staging (GLOBAL_LOAD_ASYNC_TO_LDS_B128, ASYNCcnt-tracked).
// Per-lane: 16B global (16B-aligned) -> 16B LDS (16B-aligned).
// Generic LDS pointer low 32 bits are the LDS byte address (ISA 10.2 aperture).
// ---------------------------------------------------------------------------
__device__ __forceinline__ void async_load_b128(void* lds_ptr, const void* gptr) {
    const unsigned lds_addr = (unsigned)(size_t)lds_ptr;
    const unsigned long long ga = (unsigned long long)(size_t)gptr;
    asm volatile("global_load_async_to_lds_b128 %0, %1, off"
                 :: "v"(lds_addr), "v"(ga)
                 : "memory");
}

__device__ __forceinline__ void wait_async0() {
    asm volatile("s_wait_asynccnt 0" ::: "memory");
}

// Issue all async B128 loads for one timestep's C (96x96) and F (64x96) tiles.
__device__ __forceinline__ void stage_async(const float* __restrict__ Cg,
                                            const float* __restrict__ Fg,
                                            float* sCb, float* sFb, int tid) {
    for (int idx = tid; idx < SZ_ * 24; idx += 256) {          // 9 per thread
        const int r = idx / 24, q4 = (idx % 24) * 4;
        async_load_b128(sCb + r * LDC + q4, Cg + r * SZ_ + q4);
    }
    for (int idx = tid; idx < NS_ * 24; idx += 256) {          // 6 per thread
        const int r = idx / 24, q4 = (idx % 24) * 4;
        async_load_b128(sFb + r * LDC + q4, Fg + r * SZ_ + q4);
    }
}

// ---------------------------------------------------------------------------
// LDS layout (floats), double-buffered C/F for async overlap:
// ---------------------------------------------------------------------------
#define OFF_C0  0
#define OFF_C1  (OFF_C0  + SZ_ * LDC)
#define OFF_F0  (OFF_C1  + SZ_ * LDC)
#define OFF_F1  (OFF_F0  + NS_ * LDC)
#define OFF_FV  (OFF_F1  + NS_ * LDC)
#define OFF_VM  (OFF_FV  + SZ_ * LDS65)
#define OFF_W   (OFF_VM  + NS_ * LDS65)
#define OFF_K   (OFF_W   + NC_ * LDS65)
#define OFF_L   (OFF_K   + NC_ * LDS65)
#define OFF_Q   (OFF_L   + NC_ * 33)
#define OFF_CB  (OFF_Q   + SZ_)
#define OFF_CV  (OFF_CB  + SZ_)
#define OFF_XU  (OFF_CV  + SZ_)
#define OFF_V   (OFF_XU  + SZ_)
#define OFF_KV  (OFF_V   + NS_)
#define OFF_T1  (OFF_KV  + NC_)
#define OFF_RED (OFF_T1  + NC_)
#define SMEM_FLOATS (OFF_RED + 256)

__global__ __launch_bounds__(256)
void lqr_backward_kernel(const float* __restrict__ x, const float* __restrict__ u,
                         const float* __restrict__ C, const float* __restrict__ c,
                         const float* __restrict__ F,
                         float* __restrict__ Kws, float* __restrict__ kws,
                         float* __restrict__ oldc_ws) {
    extern __shared__ float smem[];
    float* sFV  = smem + OFF_FV;
    float* sVm  = smem + OFF_VM;
    float* sW   = smem + OFF_W;
    float* sK   = smem + OFF_K;
    float* sL   = smem + OFF_L;
    float* s_q  = smem + OFF_Q;
    float* s_cb = smem + OFF_CB;
    float* s_c  = smem + OFF_CV;
    float* s_xu = smem + OFF_XU;
    float* s_v  = smem + OFF_V;
    float* s_k  = smem + OFF_KV;
    float* s_t1 = smem + OFF_T1;
    float* sred = smem + OFF_RED;

    const int b    = blockIdx.x;
    const int tid  = threadIdx.x;
    const int wave = tid >> 5;
    const int lane = tid & 31;

    // V = 0, v = 0 (scan init)
    for (int i = tid; i < NS_ * LDS65; i += 256) sVm[i] = 0.f;
    if (tid < NS_) s_v[tid] = 0.f;
    float oc_part = 0.f;   // per-thread contribution to old_cost[b]

    // Prologue: async-stage t = T-1 into buffer 0.
    {
        const size_t tb0 = (size_t)(T_ - 1) * B_ + b;
        stage_async(C + tb0 * (size_t)(SZ_ * SZ_), F + tb0 * (size_t)(NS_ * SZ_),
                    smem + OFF_C0, smem + OFF_F0, tid);
    }
    __syncthreads();

    for (int t = T_ - 1; t >= 0; --t) {
        const size_t tb  = (size_t)t * B_ + b;
        const int   cur  = (T_ - 1 - t) & 1;
        float* sC = smem + (cur ? OFF_C1 : OFF_C0);
        float* sF = smem + (cur ? OFF_F1 : OFF_F0);

        // Current buffer ready; all waves past all reads of the other buffer.
        wait_async0();
        __syncthreads();

        // Overlap: issue next timestep's tiles into the other buffer.
        if (t > 0) {
            const size_t tbn = tb - B_;
            float* sCn = smem + (cur ? OFF_C0 : OFF_C1);
            float* sFn = smem + (cur ? OFF_F0 : OFF_F1);
            stage_async(C + tbn * (size_t)(SZ_ * SZ_), F + tbn * (size_t)(NS_ * SZ_),
                        sCn, sFn, tid);
        }
        // Small per-step vectors.
        if (tid < SZ_) {
            s_c[tid]  = c[tb * SZ_ + tid];
            s_xu[tid] = (tid < NS_) ? x[tb * NS_ + tid] : u[tb * NC_ + (tid - NS_)];
        }
        __syncthreads();

        // ---- GEMM1: FV(96x64) = F^T (96x64) * V (64x64); 24 tiles ----
        for (int tt = wave; tt < 24; tt += 8) {
            const int ti = tt >> 2, tj = tt & 3;
            v8f acc = zero8();
            acc = wmma_gemm_tile(sF + 16 * ti, /*rs*/ 1, /*cs*/ LDC,   // F^T
                                 sVm + 16 * tj, LDS65, 1, NS_, acc, lane);
            store_tile(sFV + (16 * ti) * LDS65 + 16 * tj, LDS65, lane, acc);
        }
        // ---- c_back = C xu + c ; old-cost partial ; q = c_back + F^T v ----
        if (tid < SZ_) {
            float cb = s_c[tid];
            for (int j = 0; j < SZ_; ++j) cb += sC[tid * LDC + j] * s_xu[j];
            s_cb[tid] = cb;
            // 0.5 xu.(C xu) + c.xu == 0.5 xu.(c_back + c)
            oc_part += s_xu[tid] * 0.5f * (cb + s_c[tid]);
            float qv = cb;
            for (int n = 0; n < NS_; ++n) qv += sF[n * LDC + tid] * s_v[n];
            s_q[tid] = qv;
        }
        __syncthreads();

        // ---- GEMM2: Q = C + FV * F  (in-place into sC); 36 tiles ----
        for (int tt = wave; tt < 36; tt += 8) {
            const int ti = tt / 6, tj = tt % 6;
            v8f acc = load_tile(sC + (16 * ti) * LDC + 16 * tj, LDC, lane);
            acc = wmma_gemm_tile(sFV + (16 * ti) * LDS65, LDS65, 1,
                                 sF + 16 * tj, LDC, 1, NS_, acc, lane);
            store_tile(sC + (16 * ti) * LDC + 16 * tj, LDC, lane, acc);
        }
        __syncthreads();

        // ---- Cholesky of Quu (32x32, SPD) ----
        for (int idx = tid; idx < NC_ * NC_; idx += 256) {
            const int r = idx >> 5, cc = idx & 31;
            sL[r * 33 + cc] = sC[(NS_ + r) * LDC + NS_ + cc];
        }
        __syncthreads();
        for (int j = 0; j < NC_; ++j) {
            if (tid == 0) sL[j * 33 + j] = sqrtf(sL[j * 33 + j]);
            __syncthreads();
            if (tid > j && tid < NC_) sL[tid * 33 + j] /= sL[j * 33 + j];
            __syncthreads();
            if (tid > j && tid < NC_) {
                const float lij = sL[tid * 33 + j];
                for (int p = j + 1; p <= tid; ++p) sL[tid * 33 + p] -= lij * sL[p * 33 + j];
            }
            __syncthreads();
        }
        // ---- triangular solves: K = -Quu^{-1} Qux (64 cols) ; k = -Quu^{-1} qu ----
        if (tid < 65) {
            float y[NC_];
            for (int i = 0; i < NC_; ++i) {
                float s = (tid < NS_) ? sC[(NS_ + i) * LDC + tid] : s_q[NS_ + i];
                for (int p = 0; p < i; ++p) s -= sL[i * 33 + p] * y[p];
                y[i] = s / sL[i * 33 + i];
            }
            for (int i = NC_ - 1; i >= 0; --i) {
                float s = y[i];
                for (int p = i + 1; p < NC_; ++p) s -= sL[p * 33 + i] * y[p];
                y[i] = s / sL[i * 33 + i];
            }
            if (tid < NS_) {
                for (int i = 0; i < NC_; ++i) sK[i * LDS65 + tid] = -y[i];
            } else {
                for (int i = 0; i < NC_; ++i) s_k[i] = -y[i];
            }
        }
        __syncthreads();

        // ---- GEMM3: W = Quu * K (32x64); 8 tiles, one per wave. Also t1 = Quu*k ----
        {
            const int ti = wave >> 2, tj = wave & 3;
            v8f acc = zero8();
            acc = wmma_gemm_tile(sC + (NS_ + 16 * ti) * LDC + NS_, LDC, 1,
                                 sK + 16 * tj, LDS65, 1, NC_, acc, lane);
            store_tile(sW + (16 * ti) * LDS65 + 16 * tj, LDS65, lane, acc);
        }
        if (tid < NC_) {
            float s = 0.f;
            for (int p = 0; p < NC_; ++p) s += sC[(NS_ + tid) * LDC + NS_ + p] * s_k[p];
            s_t1[tid] = s;
        }
        __syncthreads();

        // ---- GEMM4: V_new = Qxx + Qxu*K + K^T*Qux + K^T*W ; 16 tiles ----
        for (int tt = wave; tt < 16; tt += 8) {
            const int ti = tt >> 2, tj = tt & 3;
            v8f acc = load_tile(sC + (16 * ti) * LDC + 16 * tj, LDC, lane);       // Qxx
            acc = wmma_gemm_tile(sC + (16 * ti) * LDC + NS_, LDC, 1,              // Qxu * K
                                 sK + 16 * tj, LDS65, 1, NC_, acc, lane);
            acc = wmma_gemm_tile(sK + 16 * ti, 1, LDS65,                          // K^T * Qux
                                 sC + NS_ * LDC + 16 * tj, LDC, 1, NC_, acc, lane);
            acc = wmma_gemm_tile(sK + 16 * ti, 1, LDS65,                          // K^T * W
                                 sW + 16 * tj, LDS65, 1, NC_, acc, lane);
            store_tile(sVm + (16 * ti) * LDS65 + 16 * tj, LDS65, lane, acc);
        }
        // ---- v_new = qx + Qxu k + K^T qu + K^T Quu k ----
        if (tid < NS_) {
            float s = s_q[tid];
            for (int m = 0; m < NC_; ++m) {
                s += sC[tid * LDC + NS_ + m] * s_k[m];
                s += sK[m * LDS65 + tid] * (s_q[NS_ + m] + s_t1[m]);
            }
            s_v[tid] = s;
        }
        // ---- spill K[t], k[t] ----
        {
            float* Kout = Kws + tb * (size_t)(NC_ * NS_);
            for (int idx = tid; idx < NC_ * NS_; idx += 256)
                Kout[idx] = sK[(idx >> 6) * LDS65 + (idx & 63)];
            if (tid < NC_) kws[tb * NC_ + tid] = s_k[tid];
        }
        __syncthreads();
    }

    // ---- reduce old_cost over the block ----
    sred[tid] = oc_part;
    __syncthreads();
    for (int s2 = 128; s2 >= 1; s2 >>= 1) {
        if (tid < s2) sred[tid] += sred[tid + s2];
        __syncthreads();
    }
    if (tid == 0) oldc_ws[b] = sred[0];
}

// ---------------------------------------------------------------------------
// Line-search rollouts: 4 iterations with alpha decay; last iteration emits
// new_x / new_u; cur_cost is the last iteration's cost. One block per b.
// Bandwidth-bound (re-streams C 4x): float4 dots + prefetch of next step's C.
// ---------------------------------------------------------------------------
__global__ __launch_bounds__(128)
void lqr_rollout_kernel(const float* __restrict__ x, const float* __restrict__ u,
                        const float* __restrict__ x_init,
                        const float* __restrict__ C, const float* __restrict__ c,
                        const float* __restrict__ F,
                        const float* __restrict__ Kws, const float* __restrict__ kws,
                        const float* __restrict__ oldc_ws,
                        float* __restrict__ out_x, float* __restrict__ out_u,
                        float* __restrict__ out_cost) {
    __shared__ float s_nx[NS_], s_dx[NS_], s_nu[NC_], s_nxu[SZ_], sred[128];

    const int b   = blockIdx.x;
    const int tid = threadIdx.x;
    const float oldc = oldc_ws[b];

    float alpha = 1.0f;
    float last_cost = 0.f;

    for (int ls = 0; ls < 4; ++ls) {
        if (tid < NS_) {
            s_nx[tid] = x_init[(size_t)b * NS_ + tid];
            s_dx[tid] = 0.f;
        }
        float cpart = 0.f;
        __syncthreads();

        for (int t = 0; t < T_; ++t) {
            const size_t tb = (size_t)t * B_ + b;
            const float* Kt = Kws + tb * (size_t)(NC_ * NS_);
            const float* kt = kws + tb * NC_;
            const float* Ft = F + tb * (size_t)(NS_ * SZ_);
            const float* Ct = C + tb * (size_t)(SZ_ * SZ_);
            const float* ct = c + tb * SZ_;
            const float* ut = u + tb * NC_;

            // Prefetch next step's C rows into cache while we compute.
            if (t + 1 < T_ && tid < SZ_) {
                const float* Cn = C + (tb + B_) * (size_t)(SZ_ * SZ_);
                __builtin_prefetch(Cn + tid * SZ_, 0, 1);
            }

            // nu = K dx + u + alpha k
            if (tid < NC_) {
                float s = ut[tid] + alpha * kt[tid];
                for (int n = 0; n < NS_; ++n) s += Kt[tid * NS_ + n] * s_dx[n];
                s_nu[tid] = s;
            }
            __syncthreads();
            if (tid < SZ_) s_nxu[tid] = (tid < NS_) ? s_nx[tid] : s_nu[tid - NS_];
            __syncthreads();

            if (ls == 3) {
                if (tid < NS_) out_x[tb * NS_ + tid] = s_nx[tid];
                if (tid < NC_) out_u[tb * NC_ + tid] = s_nu[tid];
            }
            // cost partial: 0.5 nxu.C.nxu + c.nxu  (float4 row dots)
            if (tid < SZ_) {
                float w = 0.f;
                const float4* Crow = (const float4*)(Ct + tid * SZ_);
                for (int j4 = 0; j4 < 24; ++j4) {
                    const float4 cv = Crow[j4];
                    w += cv.x * s_nxu[j4 * 4 + 0] + cv.y * s_nxu[j4 * 4 + 1] +
                         cv.z * s_nxu[j4 * 4 + 2] + cv.w * s_nxu[j4 * 4 + 3];
                }
                cpart += s_nxu[tid] * (0.5f * w + ct[tid]);
            }
            // nxt = F [nx; nu]
            float nxt = 0.f;
            if (tid < NS_) {
                const float4* Frow = (const float4*)(Ft + tid * SZ_);
                for (int j4 = 0; j4 < 24; ++j4) {
                    const float4 fv = Frow[j4];
                    nxt += fv.x * s_nxu[j4 * 4 + 0] + fv.y * s_nxu[j4 * 4 + 1] +
                           fv.z * s_nxu[j4 * 4 + 2] + fv.w * s_nxu[j4 * 4 + 3];
                }
            }
            __syncthreads();
            if (tid < NS_) {
                const int tn = (t < T_ - 1) ? (t + 1) : t;  // x_next = [x[1:], x[-1]]
                const float xn = x[((size_t)tn * B_ + b) * NS_ + tid];
                s_nx[tid] = nxt;
                s_dx[tid] = nxt - xn;
            }
            __syncthreads();
        }

        // block-reduce cost
        sred[tid] = cpart;
        __syncthreads();
        for (int s2 = 64; s2 >= 1; s2 >>= 1) {
            if (tid < s2) sred[tid] += sred[tid + s2];
            __syncthreads();
        }
        const float cost = sred[0];
        __syncthreads();
        last_cost = cost;
        if (cost > oldc) alpha *= 0.5f;   // DECAY
    }
    if (tid == 0) out_cost[b] = last_cost;
}

// ---------------------------------------------------------------------------
extern "C" void kernel_launch(void* const* d_in, const int* in_sizes, int n_in,
                              void* d_out, int out_size, void* d_ws, size_t ws_size,
                              hipStream_t stream) {
    const float* x      = (const float*)d_in[0];
    const float* u      = (const float*)d_in[1];
    const float* x_init = (const float*)d_in[2];
    const float* C      = (const float*)d_in[3];
    const float* c      = (const float*)d_in[4];
    const float* F      = (const float*)d_in[5];

    float* out      = (float*)d_out;
    float* out_x    = out;
    float* out_u    = out + (size_t)T_ * B_ * NS_;
    float* out_cost = out_u + (size_t)T_ * B_ * NC_;

    float* Kws  = (float*)d_ws;                              // T*B*NC*NS
    float* kws  = Kws + (size_t)T_ * B_ * NC_ * NS_;         // T*B*NC
    float* oldc = kws + (size_t)T_ * B_ * NC_;               // B

    const size_t smem_bytes = (size_t)SMEM_FLOATS * sizeof(float);
    lqr_backward_kernel<<<B_, 256, smem_bytes, stream>>>(x, u, C, c, F, Kws, kws, oldc);
    lqr_rollout_kernel<<<B_, 128, 0, stream>>>(x, u, x_init, C, c, F, Kws, kws, oldc,
                                               out_x, out_u, out_cost);
}